// DynamicConv_47296179864047
// MI455X (gfx1250) — compile-verified
//
#include <hip/hip_runtime.h>
#include <hip/hip_bf16.h>
#include <math.h>

// ---------------- problem constants ----------------
#define Hd   256          // hidden dim
#define Dd   64           // dynamic dim
#define Sd   49           // spatial positions (7x7)
#define NDP  32768        // ND * P
#define Pp   16384        // H * D
#define EPSL 1e-5f

typedef float v2f __attribute__((ext_vector_type(2)));
typedef float v8f __attribute__((ext_vector_type(8)));

__device__ inline v8f wmma_f32(v2f a, v2f b, v8f c) {
    // V_WMMA_F32_16X16X4_F32 : D = A(16x4) * B(4x16) + C(16x16)
    return __builtin_amdgcn_wmma_f32_16x16x4_f32(false, a, false, b, (short)0, c, false, false);
}

__device__ inline float waveReduceSum(float v) {
    #pragma unroll
    for (int m = 16; m > 0; m >>= 1) v += __shfl_xor(v, m, 32);
    return v;
}

// CDNA5 async global->LDS copy, 16B per lane, tracked by ASYNCcnt.
// Low 32 bits of a generic LDS pointer are the wave-relative LDS byte offset.
__device__ inline void async_ld_b128(void* lds, const void* g) {
    unsigned int lo = (unsigned int)(uintptr_t)lds;
    asm volatile("global_load_async_to_lds_b128 %0, %1, off"
                 :: "v"(lo), "v"(g) : "memory");
}
__device__ inline void async_wait0() {
    asm volatile("s_wait_asynccnt 0x0" ::: "memory");
}
// async loads complete in order: waiting <=3 retires everything issued
// before the most recent 3 (i.e. the previous chunk's tile is in LDS).
__device__ inline void async_wait3() {
    asm volatile("s_wait_asynccnt 0x3" ::: "memory");
}

// =====================================================================
// Generic fp32 WMMA GEMM:  C[M,N] = act(A[M,K] @ W[K,N] + bias[N])
// M mult of 64, N mult of 128, K mult of 16.  256 thr = 8 waves.
// Block tile 64x128; wave grid 4x2 -> 16x64 per wave (4 accumulators).
// Double-buffered LDS: async loads for chunk i+1 overlap compute of i.
// =====================================================================
#define SA_ST 20    // 16 + 4 pad   (64 rows)  : banks 4*l, halves +2 -> clean
#define SB_ST 144   // 128 + 16 pad (16 rows)  : halves offset 32 banks -> clean
__global__ void gemm_wmma_f32(const float* __restrict__ A,
                              const float* __restrict__ W,
                              const float* __restrict__ bias,
                              float* __restrict__ C,
                              int M, int N, int K, int act)
{
    __shared__ float sA[2][64 * SA_ST];
    __shared__ float sB[2][16 * SB_ST];

    const int tid  = threadIdx.x;
    const int wave = tid >> 5;
    const int lane = tid & 31;
    const int hf   = lane >> 4;
    const int l    = lane & 15;

    const int m_off  = (wave >> 1) * 16;
    const int n_base = (wave & 1) * 64;
    const int bm = blockIdx.y * 64;
    const int bn = blockIdx.x * 128;

    // per-thread staging coordinates (3 async b128 per thread per chunk)
    const int ar  = tid >> 2, ac4 = tid & 3;           // A tile 64x16
    const int br0 = tid >> 5, bc0 = tid & 31;          // W tile rows 0-7
    const int br1 = (tid + 256) >> 5, bc1 = bc0;       // W tile rows 8-15

    v8f acc[4];
    #pragma unroll
    for (int t = 0; t < 4; ++t) acc[t] = (v8f){};

    const int nk = K >> 4;

    // prologue: stage chunk 0 into buffer 0
    async_ld_b128(&sA[0][ar * SA_ST + ac4 * 4], A + (size_t)(bm + ar) * K + ac4 * 4);
    async_ld_b128(&sB[0][br0 * SB_ST + bc0 * 4], W + (size_t)br0 * N + bn + bc0 * 4);
    async_ld_b128(&sB[0][br1 * SB_ST + bc1 * 4], W + (size_t)br1 * N + bn + bc1 * 4);

    for (int i = 0; i < nk; ++i) {
        const int buf = i & 1;
        if (i + 1 < nk) {
            const int k0 = (i + 1) << 4;
            async_ld_b128(&sA[buf ^ 1][ar * SA_ST + ac4 * 4],
                          A + (size_t)(bm + ar) * K + k0 + ac4 * 4);
            async_ld_b128(&sB[buf ^ 1][br0 * SB_ST + bc0 * 4],
                          W + (size_t)(k0 + br0) * N + bn + bc0 * 4);
            async_ld_b128(&sB[buf ^ 1][br1 * SB_ST + bc1 * 4],
                          W + (size_t)(k0 + br1) * N + bn + bc1 * 4);
            async_wait3();          // chunk i resident, i+1 still in flight
        } else {
            async_wait0();
        }
        __syncthreads();

        #pragma unroll
        for (int kk = 0; kk < 16; kk += 4) {
            int kp = kk + hf * 2;
            v2f a; a.x = sA[buf][(m_off + l) * SA_ST + kp];
                   a.y = sA[buf][(m_off + l) * SA_ST + kp + 1];
            #pragma unroll
            for (int t = 0; t < 4; ++t) {
                int col = n_base + t * 16 + l;
                v2f b; b.x = sB[buf][kp * SB_ST + col];
                       b.y = sB[buf][(kp + 1) * SB_ST + col];
                acc[t] = wmma_f32(a, b, acc[t]);
            }
        }
        __syncthreads();   // protect buf before chunk i+2 overwrites it
    }

    #pragma unroll
    for (int t = 0; t < 4; ++t) {
        #pragma unroll
        for (int i = 0; i < 8; ++i) {
            int row = bm + m_off + i + hf * 8;
            int col = bn + n_base + t * 16 + l;
            float v = acc[t][i] + (bias ? bias[col] : 0.f);
            if (act == 1) v = fmaxf(v, 0.f);
            C[(size_t)row * N + col] = v;
        }
    }
}

// =====================================================================
// ref = sigmoid(t2 @ Wr2(256x2) + br2)   — one wave per box
// =====================================================================
__global__ void ref_head_kernel(const float* __restrict__ t2,
                                const float* __restrict__ Wr2,
                                const float* __restrict__ br2,
                                float* __restrict__ refv, int B)
{
    int wave = threadIdx.x >> 5, lane = threadIdx.x & 31;
    int b = blockIdx.x * 8 + wave;
    if (b >= B) return;
    float s0 = 0.f, s1 = 0.f;
    #pragma unroll
    for (int j = 0; j < 8; ++j) {
        int h = lane * 8 + j;
        float x = t2[(size_t)b * Hd + h];
        s0 += x * Wr2[h * 2 + 0];
        s1 += x * Wr2[h * 2 + 1];
    }
    s0 = waveReduceSum(s0); s1 = waveReduceSum(s1);
    if (lane == 0) {
        refv[b * 2 + 0] = 1.f / (1.f + __expf(-(s0 + br2[0])));
        refv[b * 2 + 1] = 1.f / (1.f + __expf(-(s1 + br2[1])));
    }
}

// =====================================================================
// px[b,h] = pos_xywh[b,h]*pos_t[b,h] * (h<128 ? ref1/bb3 : ref0/bb2)
// =====================================================================
__global__ void px_kernel(const float* __restrict__ pos_xywh,
                          const float* __restrict__ pos_t,
                          const float* __restrict__ refv,
                          const float* __restrict__ bboxes,
                          float* __restrict__ pxv)
{
    int i = blockIdx.x * 256 + threadIdx.x;
    int b = i >> 8, h = i & 255;
    float v = pos_xywh[i] * pos_t[i];
    float sc = (h < 128) ? (refv[b * 2 + 1] / bboxes[b * 4 + 3])
                         : (refv[b * 2 + 0] / bboxes[b * 4 + 2]);
    pxv[i] = v * sc;
}

// =====================================================================
// fp[b,s] = pos_local[s] * <Z[s,b,:], y[b,:]> + <Z[s,b,:], bk>
// Z laid out (S*B, 256) row-major; one wave per (b,s)
// =====================================================================
__global__ void fp_kernel(const float* __restrict__ Z,
                          const float* __restrict__ y,
                          const float* __restrict__ bk,
                          const float* __restrict__ pos_local,
                          float* __restrict__ fp, int B)
{
    int wave = threadIdx.x >> 5, lane = threadIdx.x & 31;
    int r = blockIdx.x * 8 + wave;          // r = b*49 + s
    int b = r / Sd, s = r - b * Sd;
    const float* zr = Z + (size_t)(s * B + b) * Hd + lane * 8;
    const float* yr = y + (size_t)b * Hd + lane * 8;
    const float* bkr = bk + lane * 8;
    float a0 = 0.f, a1 = 0.f;
    #pragma unroll
    for (int j = 0; j < 8; ++j) {
        float z = zr[j];
        a0 += z * yr[j];
        a1 += z * bkr[j];
    }
    a0 = waveReduceSum(a0); a1 = waveReduceSum(a1);
    if (lane == 0) fp[r] = pos_local[s] * a0 + a1;
}

// =====================================================================
// Per-box dynamic conv.  One workgroup (256 thr = 8 waves) per box.
//  F1(64x64)  = featPadded(64x256) @ param1(256x64)  (+fp) -> LN(D)+relu
//  F2(49x256) = F1(64x64) @ param2(64x256)           (pre-LN, LN later)
// =====================================================================
#define SP1_ST 80   // 64 + 16 pad   (phase-1 B tile, 64 rows)
#define SP2_ST 272  // 256 + 16 pad  (phase-2 B tile, 16 rows)
#define SF_ST  68   // 64 + 4 pad    (feat / F1 A tiles)
__global__ void perbox_kernel(const float* __restrict__ roi,     // (S,B,H)
                              const float* __restrict__ params,  // (B, 32768)
                              const float* __restrict__ fp,      // (B,S)
                              const float* __restrict__ g1,
                              const float* __restrict__ be1,
                              float* __restrict__ F2,            // (B, S*H)
                              int B)
{
    __shared__ float sP[64 * SP1_ST];   // phase2 reuses as 16 x SP2_ST (fits)
    __shared__ float sA[64 * SF_ST];    // feature chunk
    __shared__ float sF1[64 * SF_ST];   // intermediate F1

    const int b    = blockIdx.x;
    const int tid  = threadIdx.x;
    const int wave = tid >> 5;
    const int lane = tid & 31;
    const int hf   = lane >> 4;
    const int l    = lane & 15;

    const float* p1 = params + (size_t)b * NDP;        // (256 x 64) h-major
    const float* p2 = p1 + Pp;                         // (64 x 256) d-major

    // zero the pad rows of the feature tile once (never overwritten)
    for (int idx = tid; idx < 15 * 64; idx += 256) {
        int s = Sd + (idx >> 6);
        sA[s * SF_ST + (idx & 63)] = 0.f;
    }

    // ---------------- phase 1: F1 = feat @ param1 ----------------
    const int m_off = (wave >> 1) * 16;
    const int n_off = (wave & 1) * 32;
    v8f c0 = {}; v8f c1 = {};

    for (int kc = 0; kc < 4; ++kc) {                   // K=256 in chunks of 64
        // param1 chunk: 64 rows x 64 cols = 1024 float4 (contiguous source)
        #pragma unroll
        for (int q = 0; q < 4; ++q) {
            int idx = tid + q * 256;
            int r = idx >> 4, c4 = idx & 15;
            async_ld_b128(&sP[r * SP1_ST + c4 * 4],
                          p1 + kc * 4096 + r * 64 + c4 * 4);
        }
        // feat chunk: valid rows only, 49 x 16 float4
        for (int idx = tid; idx < Sd * 16; idx += 256) {
            int s = idx >> 4, c4 = idx & 15;
            async_ld_b128(&sA[s * SF_ST + c4 * 4],
                          roi + (size_t)s * B * Hd + (size_t)b * Hd + kc * 64 + c4 * 4);
        }
        async_wait0();
        __syncthreads();

        #pragma unroll
        for (int kk = 0; kk < 64; kk += 4) {
            int kp = kk + hf * 2;
            v2f a; a.x = sA[(m_off + l) * SF_ST + kp];
                   a.y = sA[(m_off + l) * SF_ST + kp + 1];
            v2f b0; b0.x = sP[kp * SP1_ST + n_off + l];
                    b0.y = sP[(kp + 1) * SP1_ST + n_off + l];
            c0 = wmma_f32(a, b0, c0);
            v2f b1; b1.x = sP[kp * SP1_ST + n_off + 16 + l];
                    b1.y = sP[(kp + 1) * SP1_ST + n_off + 16 + l];
            c1 = wmma_f32(a, b1, c1);
        }
        __syncthreads();
    }
    // + fp, stash to LDS
    #pragma unroll
    for (int i = 0; i < 8; ++i) {
        int row = m_off + i + hf * 8;
        float add = (row < Sd) ? fp[(size_t)b * Sd + row] : 0.f;
        sF1[row * SF_ST + n_off + l]      = c0[i] + add;
        sF1[row * SF_ST + n_off + 16 + l] = c1[i] + add;
    }
    __syncthreads();

    // ---------------- LayerNorm over D=64 + relu (one thread per row) ----
    if (tid < 64) {
        float m = 0.f;
        #pragma unroll
        for (int j = 0; j < 64; ++j) m += sF1[tid * SF_ST + j];
        m *= (1.f / 64.f);
        float v = 0.f;
        #pragma unroll
        for (int j = 0; j < 64; ++j) { float d = sF1[tid * SF_ST + j] - m; v += d * d; }
        v *= (1.f / 64.f);
        float rs = rsqrtf(v + EPSL);
        #pragma unroll
        for (int j = 0; j < 64; ++j) {
            float x = (sF1[tid * SF_ST + j] - m) * rs * g1[j] + be1[j];
            sF1[tid * SF_ST + j] = fmaxf(x, 0.f);
        }
    }
    __syncthreads();

    // ---------------- phase 2: F2 = F1(64x64) @ param2(64x256) ----------
    const int n_base = (wave & 1) * 128;               // 8 col tiles of 16
    v8f acc[8];
    #pragma unroll
    for (int t = 0; t < 8; ++t) acc[t] = (v8f){};

    for (int kc = 0; kc < 4; ++kc) {                   // K=64 in chunks of 16
        // param2 chunk: 16 rows x 256 cols = 1024 float4 (contiguous source)
        #pragma unroll
        for (int q = 0; q < 4; ++q) {
            int idx = tid + q * 256;
            int r = idx >> 6, c4 = idx & 63;
            async_ld_b128(&sP[r * SP2_ST + c4 * 4],
                          p2 + kc * 4096 + r * 256 + c4 * 4);
        }
        async_wait0();
        __syncthreads();

        #pragma unroll
        for (int kk = 0; kk < 16; kk += 4) {
            int kp = kk + hf * 2;
            int kg = kc * 16 + kp;
            v2f a; a.x = sF1[(m_off + l) * SF_ST + kg];
                   a.y = sF1[(m_off + l) * SF_ST + kg + 1];
            #pragma unroll
            for (int t = 0; t < 8; ++t) {
                int col = n_base + t * 16 + l;
                v2f bb; bb.x = sP[kp * SP2_ST + col];
                        bb.y = sP[(kp + 1) * SP2_ST + col];
                acc[t] = wmma_f32(a, bb, acc[t]);
            }
        }
        __syncthreads();
    }
    // write pre-LN F2 (valid rows only)
    #pragma unroll
    for (int t = 0; t < 8; ++t) {
        #pragma unroll
        for (int i = 0; i < 8; ++i) {
            int row = m_off + i + hf * 8;
            if (row < Sd) {
                int col = n_base + t * 16 + l;
                F2[(size_t)b * (Sd * Hd) + row * Hd + col] = acc[t][i];
            }
        }
    }
}

// =====================================================================
// LayerNorm over 256 + relu — one wave per row
// =====================================================================
__global__ void ln256_relu(const float* __restrict__ in,
                           float* __restrict__ out,
                           const float* __restrict__ g,
                           const float* __restrict__ be)
{
    int wave = threadIdx.x >> 5, lane = threadIdx.x & 31;
    size_t r = (size_t)blockIdx.x * 8 + wave;
    const float* row = in + r * Hd + lane * 8;
    float x[8], m = 0.f;
    #pragma unroll
    for (int j = 0; j < 8; ++j) { x[j] = row[j]; m += x[j]; }
    m = waveReduceSum(m) * (1.f / 256.f);
    float v = 0.f;
    #pragma unroll
    for (int j = 0; j < 8; ++j) { float d = x[j] - m; v += d * d; }
    v = waveReduceSum(v) * (1.f / 256.f);
    float rs = rsqrtf(v + EPSL);
    float* orow = out + r * Hd + lane * 8;
    #pragma unroll
    for (int j = 0; j < 8; ++j) {
        int h = lane * 8 + j;
        orow[j] = fmaxf((x[j] - m) * rs * g[h] + be[h], 0.f);
    }
}

// =====================================================================
extern "C" void kernel_launch(void* const* d_in, const int* in_sizes, int n_in,
                              void* d_out, int out_size, void* d_ws, size_t ws_size,
                              hipStream_t stream)
{
    const int B = 4800;
    const float* pro      = (const float*)d_in[0];   // (1,B,H)
    const float* roi      = (const float*)d_in[1];   // (S,B,H)
    const float* pos_xywh = (const float*)d_in[2];   // (1,B,H)
    const float* pos_img  = (const float*)d_in[3];   // (S,B,H)
    const float* pos_loc  = (const float*)d_in[4];   // (7,7)
    const float* bboxes   = (const float*)d_in[5];   // (B,4)
    const float* Wd   = (const float*)d_in[6];  const float* bd   = (const float*)d_in[7];
    const float* Wqs1 = (const float*)d_in[8];  const float* bqs1 = (const float*)d_in[9];
    const float* Wqs2 = (const float*)d_in[10]; const float* bqs2 = (const float*)d_in[11];
    const float* Wr1  = (const float*)d_in[12]; const float* br1  = (const float*)d_in[13];
    const float* Wr2  = (const float*)d_in[14]; const float* br2  = (const float*)d_in[15];
    const float* Wq   = (const float*)d_in[16]; const float* bq   = (const float*)d_in[17];
    const float* Wk   = (const float*)d_in[18]; const float* bk   = (const float*)d_in[19];
    const float* g1   = (const float*)d_in[20]; const float* be1  = (const float*)d_in[21];
    const float* g2   = (const float*)d_in[22]; const float* be2  = (const float*)d_in[23];
    const float* g3   = (const float*)d_in[24]; const float* be3  = (const float*)d_in[25];
    const float* Wo   = (const float*)d_in[26]; const float* bo   = (const float*)d_in[27];
    float* out = (float*)d_out;

    // ---- workspace carve-up (floats) ----
    float* w = (float*)d_ws;
    size_t off = 0;
    float* t1     = w + off; off += (size_t)B * Hd;
    float* pos_t  = w + off; off += (size_t)B * Hd;
    float* t2     = w + off; off += (size_t)B * Hd;
    float* refv   = w + off; off += (size_t)B * 2;
    float* pxv    = w + off; off += (size_t)B * Hd;
    float* yv     = w + off; off += (size_t)B * Hd;
    float* fpv    = w + off; off += (size_t)B * Sd;
    float* outpre = w + off; off += (size_t)B * Hd;
    float* F2     = w + off; off += (size_t)B * Sd * Hd;
    float* params = w + off; off += (size_t)B * NDP;
    float* Z      = params;   // Z (S*B,256) is dead before params is written

    dim3 blk(256);
    const int SB = Sd * B;   // 235200

    // query_scale MLP: t1 = relu(pro@Wqs1+b), pos_t = t1@Wqs2+b
    gemm_wmma_f32<<<dim3(Hd / 128, B / 64), blk, 0, stream>>>(pro, Wqs1, bqs1, t1, B, Hd, Hd, 1);
    gemm_wmma_f32<<<dim3(Hd / 128, B / 64), blk, 0, stream>>>(t1, Wqs2, bqs2, pos_t, B, Hd, Hd, 0);
    // ref_point head: t2 = relu(pro@Wr1+b); ref = sigmoid(t2@Wr2+b)
    gemm_wmma_f32<<<dim3(Hd / 128, B / 64), blk, 0, stream>>>(pro, Wr1, br1, t2, B, Hd, Hd, 1);
    ref_head_kernel<<<B / 8, blk, 0, stream>>>(t2, Wr2, br2, refv, B);
    // px vector, y = px@Wk
    px_kernel<<<B * Hd / 256, blk, 0, stream>>>(pos_xywh, pos_t, refv, bboxes, pxv);
    gemm_wmma_f32<<<dim3(Hd / 128, B / 64), blk, 0, stream>>>(pxv, Wk, nullptr, yv, B, Hd, Hd, 0);
    // Z = pos_img @ Wq + bq   (S*B x 256)
    gemm_wmma_f32<<<dim3(Hd / 128, SB / 64), blk, 0, stream>>>(pos_img, Wq, bq, Z, SB, Hd, Hd, 0);
    // fp[b,s]
    fp_kernel<<<SB / 8, blk, 0, stream>>>(Z, yv, bk, pos_loc, fpv, B);
    // big GEMM: parameters = pro @ Wd + bd   (overwrites Z — stream-ordered)
    gemm_wmma_f32<<<dim3(NDP / 128, B / 64), blk, 0, stream>>>(pro, Wd, bd, params, B, NDP, Hd, 0);
    // per-box dynamic conv
    perbox_kernel<<<B, blk, 0, stream>>>(roi, params, fpv, g1, be1, F2, B);
    // LN over H + relu, in place on F2
    ln256_relu<<<SB / 8, blk, 0, stream>>>(F2, F2, g2, be2);
    // out head: outpre = F2(B x 12544) @ Wo + bo ; final LN+relu -> d_out
    gemm_wmma_f32<<<dim3(Hd / 128, B / 64), blk, 0, stream>>>(F2, Wo, bo, outpre, B, Hd, Sd * Hd, 0);
    ln256_relu<<<B / 8, blk, 0, stream>>>(outpre, out, g3, be3);
}